// Router_61263413510548
// MI455X (gfx1250) — compile-verified
//
#include <hip/hip_runtime.h>

typedef float v2f __attribute__((ext_vector_type(2)));
typedef float v8f __attribute__((ext_vector_type(8)));

#define NUM_E   256
#define DIM_K   1024
#define TOPK    8
#define TPB     32          // tokens per block
#define SSTR    260         // LDS score row stride (bank-conflict-free halves)
#define NEG_INF (-3.402823466e+38f)

__device__ __forceinline__ void topk_insert(float v, int idx, float (&bv)[TOPK], int (&bi)[TOPK]) {
    if (v > bv[TOPK - 1]) {
        bv[TOPK - 1] = v;
        bi[TOPK - 1] = idx;
#pragma unroll
        for (int j = TOPK - 1; j >= 1; --j) {
            if (bv[j] > bv[j - 1]) {
                float tv = bv[j - 1]; bv[j - 1] = bv[j]; bv[j] = tv;
                int   ti = bi[j - 1]; bi[j - 1] = bi[j]; bi[j] = ti;
            }
        }
    }
}

__global__ __launch_bounds__(256) void
router_kernel(const float* __restrict__ x, const float* __restrict__ w,
              const float* __restrict__ bias, float* __restrict__ out_w,
              int* __restrict__ out_i) {
    __shared__ float s_scores[TPB * SSTR];   // sigmoid scores, 32 tokens x 256 experts
    __shared__ float s_bias[NUM_E];
    __shared__ float s_cv[TPB * 64];         // merge candidates (values = score+bias)
    __shared__ int   s_ci[TPB * 64];         // merge candidates (expert index)

    const int tid = threadIdx.x;
    s_bias[tid] = bias[tid];                 // 256 threads == 256 experts

    // ---------------- GEMM phase: D = x @ W^T via V_WMMA_F32_16X16X4_F32 ----
    const int wave = tid >> 5;               // 0..7
    const int lane = tid & 31;
    const int half = lane >> 4;              // 0: K={0,1}, 1: K={2,3}
    const int r    = lane & 15;
    const int mt   = wave >> 2;              // M-tile (16 tokens each)
    const int nb   = (wave & 3) * 64;        // expert base for this wave

    const int tok0 = blockIdx.x * TPB + mt * 16;

    const float* pa  = x + (size_t)(tok0 + r) * DIM_K + 2 * half;
    const float* pb0 = w + (size_t)(nb +  0 + r) * DIM_K + 2 * half;
    const float* pb1 = w + (size_t)(nb + 16 + r) * DIM_K + 2 * half;
    const float* pb2 = w + (size_t)(nb + 32 + r) * DIM_K + 2 * half;
    const float* pb3 = w + (size_t)(nb + 48 + r) * DIM_K + 2 * half;

    v8f acc0 = {}, acc1 = {}, acc2 = {}, acc3 = {};

#pragma unroll 1
    for (int kk = 0; kk < DIM_K; kk += 64) {
        if (kk + 64 < DIM_K) {               // prefetch next K chunk (speculative)
            __builtin_prefetch(pa  + kk + 64, 0, 1);
            __builtin_prefetch(pb0 + kk + 64, 0, 1);
            __builtin_prefetch(pb1 + kk + 64, 0, 1);
            __builtin_prefetch(pb2 + kk + 64, 0, 1);
            __builtin_prefetch(pb3 + kk + 64, 0, 1);
        }
#pragma unroll
        for (int k4 = 0; k4 < 64; k4 += 4) {
            const int k = kk + k4;
            v2f a  = *(const v2f*)(pa  + k);
            v2f b0 = *(const v2f*)(pb0 + k);
            v2f b1 = *(const v2f*)(pb1 + k);
            v2f b2 = *(const v2f*)(pb2 + k);
            v2f b3 = *(const v2f*)(pb3 + k);
            acc0 = __builtin_amdgcn_wmma_f32_16x16x4_f32(false, a, false, b0, (short)0, acc0, false, false);
            acc1 = __builtin_amdgcn_wmma_f32_16x16x4_f32(false, a, false, b1, (short)0, acc1, false, false);
            acc2 = __builtin_amdgcn_wmma_f32_16x16x4_f32(false, a, false, b2, (short)0, acc2, false, false);
            acc3 = __builtin_amdgcn_wmma_f32_16x16x4_f32(false, a, false, b3, (short)0, acc3, false, false);
        }
    }

    // sigmoid + scatter to LDS. C layout: VGPR j, lanes0-15 -> M=j, lanes16-31 -> M=j+8, N=r
#pragma unroll
    for (int j = 0; j < 8; ++j) {
        const int m = mt * 16 + j + 8 * half;
        float* row = &s_scores[m * SSTR];
        row[nb +  0 + r] = 1.0f / (1.0f + __expf(-acc0[j]));
        row[nb + 16 + r] = 1.0f / (1.0f + __expf(-acc1[j]));
        row[nb + 32 + r] = 1.0f / (1.0f + __expf(-acc2[j]));
        row[nb + 48 + r] = 1.0f / (1.0f + __expf(-acc3[j]));
    }

    __syncthreads();

    // ---------------- Top-K phase A: 8 threads/token, 32 experts each --------
    {
        const int t   = tid >> 3;            // token 0..31
        const int sub = tid & 7;             // expert sub-range
        float bv[TOPK]; int bi[TOPK];
#pragma unroll
        for (int j = 0; j < TOPK; ++j) { bv[j] = NEG_INF; bi[j] = -1; }

        const float* srow = &s_scores[t * SSTR + sub * 32];
        const float* brow = &s_bias[sub * 32];
#pragma unroll 4
        for (int i = 0; i < 32; ++i)
            topk_insert(srow[i] + brow[i], sub * 32 + i, bv, bi);

#pragma unroll
        for (int j = 0; j < TOPK; ++j) {
            s_cv[t * 64 + sub * TOPK + j] = bv[j];
            s_ci[t * 64 + sub * TOPK + j] = bi[j];
        }
    }

    __syncthreads();

    // ---------------- Top-K phase B: merge 64 candidates, gather, normalize --
    if (tid < TPB) {
        const int t = tid;
        float bv[TOPK]; int bi[TOPK];
#pragma unroll
        for (int j = 0; j < TOPK; ++j) { bv[j] = NEG_INF; bi[j] = -1; }

#pragma unroll 4
        for (int c = 0; c < 64; ++c)
            topk_insert(s_cv[t * 64 + c], s_ci[t * 64 + c], bv, bi);

        float ws[TOPK];
        float sum = 0.0f;
#pragma unroll
        for (int j = 0; j < TOPK; ++j) {
            ws[j] = s_scores[t * SSTR + bi[j]];   // original sigmoid score
            sum += ws[j];
        }
        const float scale = 2.5f / sum;
        const int g = blockIdx.x * TPB + t;
#pragma unroll
        for (int j = 0; j < TOPK; ++j) {
            out_w[g * TOPK + j] = ws[j] * scale;
            out_i[g * TOPK + j] = bi[j];
        }
    }
}

extern "C" void kernel_launch(void* const* d_in, const int* in_sizes, int n_in,
                              void* d_out, int out_size, void* d_ws, size_t ws_size,
                              hipStream_t stream) {
    const float* x    = (const float*)d_in[0];
    const float* wgt  = (const float*)d_in[1];
    const float* bias = (const float*)d_in[2];
    const int M = in_sizes[0] / DIM_K;                 // 32768 tokens

    float* out_w = (float*)d_out;                      // weights [M,8] f32
    int*   out_i = (int*)((float*)d_out + (size_t)M * TOPK); // indices [M,8] i32

    dim3 grid(M / TPB);
    dim3 block(256);
    router_kernel<<<grid, block, 0, stream>>>(x, wgt, bias, out_w, out_i);
}